// MeDILCausalModel_89223650607714
// MI455X (gfx1250) — compile-verified
//
#include <hip/hip_runtime.h>

// Problem constants (from reference): B=8192, L=128, O=512, K=8, H=20
#define B_ 8192
#define L_ 128
#define O_ 512
#define K_ 8
#define H_ 20
#define HS_ 36                      // h-buffer row stride (floats): 8*36 % 64 == 32 -> conflict-free
#define SPLIT_ 8                    // grid.y batch split
#define RPB_ (B_ / SPLIT_)          // rows per block = 1024
#define TILES_PER_WAVE_ (RPB_ / (8 * 16))  // 8 waves * 16 rows -> 8 iterations

#define KA1_ 12                     // layer-1 K padded: 9 inputs + bias row + 2 zero rows
#define KA2_ 24                     // layer-2 K padded: 20 inputs + bias row + 3 zero rows

typedef __attribute__((ext_vector_type(2))) float v2f;
typedef __attribute__((ext_vector_type(4))) float v4f;
typedef __attribute__((ext_vector_type(8))) float v8f;

// fp32 WMMA: D(16x16,f32) = A(16x4,f32) * B(4x16,f32) + C   -> v_wmma_f32_16x16x4_f32
__device__ __forceinline__ v8f wmma4(v2f a, v2f b, v8f c) {
  return __builtin_amdgcn_wmma_f32_16x16x4_f32(
      /*neg_a=*/false, a, /*neg_b=*/false, b,
      /*c_mod=*/(short)0, c, /*reuse_a=*/false, /*reuse_b=*/false);
}

__device__ __forceinline__ float fast_tanh(float v) {
#if __has_builtin(__builtin_amdgcn_tanhf)
  return __builtin_amdgcn_tanhf(v);       // v_tanh_f32 (gfx1250 trans op)
#else
  float e = __expf(2.0f * v);
  return (e - 1.0f) * __builtin_amdgcn_rcpf(e + 1.0f);
#endif
}

__global__ __launch_bounds__(256) void medil_mlp_wmma_kernel(
    const float* __restrict__ x,      // (B, L)
    const float* __restrict__ noise,  // (B, O)
    const int*   __restrict__ cidx,   // (O, K)
    const float* __restrict__ W1,     // (O, K+1, H)
    const float* __restrict__ b1,     // (O, H)
    const float* __restrict__ W2,     // (O, H, H)
    const float* __restrict__ b2,     // (O, H)
    const float* __restrict__ W3,     // (O, H)
    const float* __restrict__ b3,     // (O,)
    float*       __restrict__ out)    // (B, O)
{
  __shared__ float sW1e[KA1_ * H_];      // augmented: rows 0..8=W1, 9=b1, 10..11=0
  __shared__ float sW2e[KA2_ * H_];      // augmented: rows 0..19=W2, 20=b2, 21..23=0
  __shared__ float sW3[H_];
  __shared__ float sb3s;
  __shared__ float hbuf[8][16 * HS_];    // per-wave 16x(20 valid of 32) activation scratch

  const int o   = blockIdx.x;
  const int tid = threadIdx.x;

  // ---- stage augmented weights into LDS (coalesced, division-free) ----
  // sW1e flat: [0,180) = W1 block, [180,200) = b1, [200,240) = 0
  for (int t = tid; t < KA1_ * H_; t += 256) {
    float v = 0.0f;
    if (t < (K_ + 1) * H_)            v = W1[o * (K_ + 1) * H_ + t];
    else if (t < (K_ + 2) * H_)       v = b1[o * H_ + (t - (K_ + 1) * H_)];
    sW1e[t] = v;
  }
  // sW2e flat: [0,400) = W2 block, [400,420) = b2, [420,480) = 0
  for (int t = tid; t < KA2_ * H_; t += 256) {
    float v = 0.0f;
    if (t < H_ * H_)                  v = W2[o * H_ * H_ + t];
    else if (t < (H_ + 1) * H_)       v = b2[o * H_ + (t - H_ * H_)];
    sW2e[t] = v;
  }
  if (tid < H_) sW3[tid] = W3[o * H_ + tid];
  if (tid == 0) sb3s = b3[o];
  __syncthreads();

  const int lane = tid & 31;
  const int wave = tid >> 5;
  const int hi   = lane >> 4;   // 0: lanes 0-15, 1: lanes 16-31
  const int ln   = lane & 15;

  // gather indices for this o (wave-uniform -> SGPRs)
  const int ix0 = cidx[o * K_ + 0], ix1 = cidx[o * K_ + 1];
  const int ix2 = cidx[o * K_ + 2], ix3 = cidx[o * K_ + 3];
  const int ix4 = cidx[o * K_ + 4], ix5 = cidx[o * K_ + 5];
  const int ix6 = cidx[o * K_ + 6], ix7 = cidx[o * K_ + 7];

  // ---- loop-invariant B fragments (lane = N column, regs hold K slice) ----
  // B layout for 16x16x4: lanes0-15 hold K={0,1}, lanes16-31 hold K={2,3}; N = lane%16
  v2f fW1[3][2];
#pragma unroll
  for (int g = 0; g < 3; ++g) {
#pragma unroll
    for (int n = 0; n < 2; ++n) {
      const int N  = n * 16 + ln;
      const int k0 = g * 4 + hi * 2;
      fW1[g][n].x = (N < H_) ? sW1e[k0 * H_ + N]       : 0.0f;
      fW1[g][n].y = (N < H_) ? sW1e[(k0 + 1) * H_ + N] : 0.0f;
    }
  }
  v2f fW2[6][2];
#pragma unroll
  for (int g = 0; g < 6; ++g) {
#pragma unroll
    for (int n = 0; n < 2; ++n) {
      const int N  = n * 16 + ln;
      const int k0 = g * 4 + hi * 2;
      fW2[g][n].x = (N < H_) ? sW2e[k0 * H_ + N]       : 0.0f;
      fW2[g][n].y = (N < H_) ? sW2e[(k0 + 1) * H_ + N] : 0.0f;
    }
  }
  const float b3r = sb3s;

  // W3 hoisted into registers (used by lanes 0-15 in layer 3)
  v4f w3r[5];
#pragma unroll
  for (int q = 0; q < 5; ++q) w3r[q] = *(const v4f*)&sW3[q * 4];

  // ---- loop-invariant per-lane gather columns ----
  const int cA = hi ? ix2 : ix0;                        // K = 1 / 3
  const int cB = hi ? ix5 : ix3;                        // K = 4 / 6
  const int cC = hi ? ix6 : ix4;                        // K = 5 / 7

  // ---- incremental 32-bit element offsets (row advance per iter is constant) ----
  const int brow0 = blockIdx.y * RPB_ + wave * 16 + ln;
  unsigned offA = (unsigned)(brow0 * L_ + cA);
  unsigned offB = (unsigned)(brow0 * L_ + cB);
  unsigned offC = (unsigned)(brow0 * L_ + cC);
  unsigned off7 = (unsigned)(brow0 * L_ + ix7);
  const unsigned stepX = 128u * L_;                     // 8 waves * 16 rows
  unsigned offOut = (unsigned)(brow0 * O_ + o);
  const unsigned stepOut = 128u * O_;
  // divergent-base operand (noise for lo lanes, x-gather for hi lanes): one pointer
  const float* p0 = hi ? (x + (brow0 * L_ + ix1)) : (noise + (brow0 * O_ + o));
  const size_t step0 = hi ? (size_t)(128 * L_) : (size_t)(128 * O_);

  float* hb = hbuf[wave];

#pragma unroll 1
  for (int it = 0; it < TILES_PER_WAVE_; ++it) {
    // ---- layer 1 A fragments (branch-free): K rows 9..11 of B are {b1,0,0} ----
    v2f a0, a1, a2;
    a0.x = *p0;                               // K = 0 (noise, lo) / 2 (hi)
    a0.y = x[offA];                           // K = 1 / 3
    a1.x = x[offB];                           // K = 4 / 6
    a1.y = x[offC];                           // K = 5 / 7
    a2.x = x[off7];                           // K = 8 real / 10 (zero B row)
    a2.y = 1.0f;                              // K = 9 bias row / 11 (zero B row)

    // ---- layer 1 (+bias in K=9): 16x20 = [inp,1](16x10) @ [W1;b1](10x20) ----
    v8f c0 = {}, c1 = {};
    c0 = wmma4(a0, fW1[0][0], c0);  c1 = wmma4(a0, fW1[0][1], c1);
    c0 = wmma4(a1, fW1[1][0], c0);  c1 = wmma4(a1, fW1[1][1], c1);
    c0 = wmma4(a2, fW1[2][0], c0);  c1 = wmma4(a2, fW1[2][1], c1);

    // tanh, write h1 to per-wave LDS (unconditional; cols 20..31 are dead zeros)
#pragma unroll
    for (int j = 0; j < 8; ++j) {
      const int m = j + 8 * hi;
      hb[m * HS_ + ln]      = fast_tanh(c0[j]);
      hb[m * HS_ + 16 + ln] = fast_tanh(c1[j]);
    }

    // ---- layer 2 (+bias in K=20): 5 data K-steps + 1 bias K-step, 2 N tiles ----
    v8f d0 = {}, d1 = {};
#pragma unroll
    for (int g = 0; g < 5; ++g) {
      const int k0 = g * 4 + hi * 2;
      const v2f a = *(const v2f*)&hb[ln * HS_ + k0];   // ds_load_b64, transposed read
      d0 = wmma4(a, fW2[g][0], d0);
      d1 = wmma4(a, fW2[g][1], d1);
    }
    {
      const v2f aone = {1.0f, 1.0f};                   // bias row K=20 (others zero)
      d0 = wmma4(aone, fW2[5][0], d0);
      d1 = wmma4(aone, fW2[5][1], d1);
    }

    // tanh, write h2 back to LDS (unconditional)
#pragma unroll
    for (int j = 0; j < 8; ++j) {
      const int m = j + 8 * hi;
      hb[m * HS_ + ln]      = fast_tanh(d0[j]);
      hb[m * HS_ + 16 + ln] = fast_tanh(d1[j]);
    }

    // ---- layer 3: out[b,o] = h2 . W3 + b3 (lanes 0-15, one row each) ----
    if (hi == 0) {
      const v4f* hr = (const v4f*)&hb[ln * HS_];      // 144*ln bytes: 16B aligned
      float acc = b3r;
#pragma unroll
      for (int q = 0; q < 5; ++q) {                   // 5 x ds_load_b128
        const v4f hv = hr[q];
        const v4f wv = w3r[q];
        acc += hv.x * wv.x + hv.y * wv.y + hv.z * wv.z + hv.w * wv.w;
      }
      out[offOut] = acc;
    }

    // advance incremental offsets (constant strides)
    p0     += step0;
    offA   += stepX;
    offB   += stepX;
    offC   += stepX;
    off7   += stepX;
    offOut += stepOut;
  }
}

extern "C" void kernel_launch(void* const* d_in, const int* in_sizes, int n_in,
                              void* d_out, int out_size, void* d_ws, size_t ws_size,
                              hipStream_t stream) {
  (void)in_sizes; (void)n_in; (void)out_size; (void)d_ws; (void)ws_size;
  const float* x     = (const float*)d_in[0];
  const float* noise = (const float*)d_in[1];
  const int*   cidx  = (const int*)  d_in[2];
  const float* W1    = (const float*)d_in[3];
  const float* b1    = (const float*)d_in[4];
  const float* W2    = (const float*)d_in[5];
  const float* b2    = (const float*)d_in[6];
  const float* W3    = (const float*)d_in[7];
  const float* b3    = (const float*)d_in[8];
  float* out = (float*)d_out;

  dim3 grid(O_, SPLIT_, 1);   // 512 x 8 blocks
  dim3 block(256, 1, 1);      // 8 wave32 per block
  hipLaunchKernelGGL(medil_mlp_wmma_kernel, grid, block, 0, stream,
                     x, noise, cidx, W1, b1, W2, b2, W3, b3, out);
}